// ConformerB2_10282151707576
// MI455X (gfx1250) — compile-verified
//
#include <hip/hip_runtime.h>
#include <hip/hip_bf16.h>
#include <math.h>

constexpr int kE = 40, kDepth = 6, kNH = 5, kFF = 160, kHid = 64, kNC = 4, kWS = 8;
constexpr int kB = 32, kN = 200, kXH = 22, kXW = 4096, kK1 = 25, kCW = 4055;
constexpr int kNp = 208;  // 13 row-tiles of 16

typedef __attribute__((ext_vector_type(2))) float v2f;
typedef __attribute__((ext_vector_type(8))) float v8f;

__device__ __forceinline__ float elu1(float x) { return x > 0.f ? x : (expf(x) - 1.f); }

// ---------------- Kernel 1: fold conv1 (1x25) + conv2 (22x1) into one 22x25 conv ----------------
__global__ void k_fold(const float* __restrict__ w1, const float* __restrict__ b1,
                       const float* __restrict__ w2, const float* __restrict__ b2,
                       float* __restrict__ weff, float* __restrict__ beff) {
  int idx = blockIdx.x * blockDim.x + threadIdx.x;
  if (idx < 40 * 22 * 25) {
    int k = idx % 25, h = (idx / 25) % 22, o = idx / (25 * 22);
    float s = 0.f;
    for (int i = 0; i < 40; ++i) s += w2[(o * 40 + i) * 22 + h] * w1[i * 25 + k];
    weff[idx] = s;
  }
  if (idx < 40) {
    float s = b2[idx];
    for (int i = 0; i < 40; ++i) {
      float bi = b1[i];
      for (int h = 0; h < 22; ++h) s += bi * w2[(idx * 40 + i) * 22 + h];
    }
    beff[idx] = s;
  }
}

// ---------------- Kernel 2: fused conv + BN + ELU, only the 4055 positions pooling needs ----------------
__global__ void k_conv(const float* __restrict__ x, const float* __restrict__ weff,
                       const float* __restrict__ beff,
                       const float* __restrict__ bn_g, const float* __restrict__ bn_b,
                       const float* __restrict__ bn_m, const float* __restrict__ bn_v,
                       float* __restrict__ cact) {
  long long idx = (long long)blockIdx.x * blockDim.x + threadIdx.x;
  long long tot = (long long)kB * 40 * kCW;
  if (idx >= tot) return;
  int w = (int)(idx % kCW);
  int o = (int)((idx / kCW) % 40);
  int b = (int)(idx / ((long long)kCW * 40));
  const float* xb = x + (long long)b * kXH * kXW;
  const float* wgt = weff + o * (22 * 25);
  float s = beff[o];
  for (int h = 0; h < kXH; ++h) {
    const float* xr = xb + h * kXW + w;
    const float* wr = wgt + h * kK1;
    for (int k = 0; k < kK1; ++k) s += xr[k] * wr[k];
  }
  float sc = bn_g[o] * rsqrtf(bn_v[o] + 1e-5f);
  s = (s - bn_m[o]) * sc + bn_b[o];
  cact[idx] = elu1(s);
}

// ---------------- Kernel 3: mean-pool (75/stride 20) + 40x40 projection -> tokens ----------------
__global__ void k_pool_proj(const float* __restrict__ cact, const float* __restrict__ pw,
                            const float* __restrict__ pb, float* __restrict__ tok) {
  int b = blockIdx.x / kN;
  int p = blockIdx.x % kN;
  __shared__ float pooled[40];
  int tid = threadIdx.x;
  if (tid < 40) {
    const float* src = cact + (long long)(b * 40 + tid) * kCW + p * 20;
    float s = 0.f;
    for (int j = 0; j < 75; ++j) s += src[j];
    pooled[tid] = s * (1.f / 75.f);
  }
  __syncthreads();
  if (tid < 40) {
    float s = pb[tid];
    for (int o = 0; o < 40; ++o) s += pooled[o] * pw[o * 40 + tid];
    tok[(long long)(b * kN + p) * kE + tid] = s;
  }
}

// ---------------- wave-level f32 WMMA tile GEMM:  C[MxN] (+)= A[MxK] * B[KxN] + bias ----------------
// All fragment loads are UNCONDITIONAL:
//  * A rows >= real M over-read adjacent LDS -> garbage only feeds C rows we discard / pad rows.
//  * B cols >= Ncols are clamped to Ncols-1 -> garbage only feeds C cols we never store.
// A pair (K, K+1) is contiguous -> single ds_load_b64 per step.
__device__ __forceinline__ void wave_gemm(const float* A, int lda,
                                          const float* Bm, int ldb, int Ncols,
                                          const float* bias,
                                          float* C, int ldc,
                                          int Mtiles, int Kdim,
                                          bool accum, bool rowcheck, int Crows) {
  const int tid = threadIdx.x;
  const int lane = tid & 31;
  const int wave = tid >> 5;
  const int nwaves = blockDim.x >> 5;
  const int ntiles = (Ncols + 15) >> 4;
  const int m16 = lane & 15;
  const int kb = (lane >> 4) << 1;     // K pair held by this half-wave: 0 or 2
  const int rbase = (lane >> 4) << 3;  // C rows 0..7 vs 8..15
  for (int t = wave; t < Mtiles * ntiles; t += nwaves) {
    int mt = t / ntiles, nt = t % ntiles;
    int m0 = mt << 4, n0 = nt << 4;
    int arow = m0 + m16;
    int bcol = n0 + m16;
    int bcolc = bcol < Ncols ? bcol : (Ncols - 1);
    const float* ap = A + arow * lda;
    v8f acc = {0.f, 0.f, 0.f, 0.f, 0.f, 0.f, 0.f, 0.f};
    for (int k0 = 0; k0 < Kdim; k0 += 4) {
      int ka = k0 + kb;
      v2f av = *(const v2f*)(ap + ka);
      v2f bv;
      bv.x = Bm[ka * ldb + bcolc];
      bv.y = Bm[(ka + 1) * ldb + bcolc];
      acc = __builtin_amdgcn_wmma_f32_16x16x4_f32(false, av, false, bv, (short)0, acc,
                                                  false, false);
    }
    if (bcol < Ncols) {
      float bc = (bias != nullptr) ? bias[bcol] : 0.f;
      for (int r = 0; r < 8; ++r) {
        int row = m0 + rbase + r;
        if (!rowcheck || row < Crows) {
          float v = acc[r] + bc;
          float* cp = C + row * ldc + bcol;
          *cp = accum ? (*cp + v) : v;
        }
      }
    }
  }
}

// ---------------- Kernel 4: whole transformer (6 layers) + pooling head, one block per batch elem ----------------
__global__ void __launch_bounds__(256) k_former(
    const float* __restrict__ tok,
    const float* __restrict__ ln1_g, const float* __restrict__ ln1_b,
    const float* __restrict__ wq, const float* __restrict__ bq,
    const float* __restrict__ wk, const float* __restrict__ bk,
    const float* __restrict__ wv, const float* __restrict__ bvv,
    const float* __restrict__ wo, const float* __restrict__ bo,
    const float* __restrict__ ln2_g, const float* __restrict__ ln2_b,
    const float* __restrict__ fw1, const float* __restrict__ fb1,
    const float* __restrict__ fw2, const float* __restrict__ fb2,
    const float* __restrict__ attn_w, const float* __restrict__ attn_b,
    const float* __restrict__ cw1, const float* __restrict__ cb1,
    const float* __restrict__ cw2, const float* __restrict__ cb2,
    float* __restrict__ out) {
  // Exactly 163,840 B of LDS. Q/K/V are padded to 208 rows (no row checks on their stores);
  // H1 (FF hidden, 208x80) reuses the Q+K region exactly.
  __shared__ float smem[40960];
  float* T = smem;            // 8000: residual stream [200][40] (row-checked accum stores)
  float* Y = smem + 8000;     // 8000: LN out / attention O / head scratch
  float* Qm = smem + 16000;   // 8320 = 208x40
  float* Km = smem + 24320;   // 8320
  float* Vm = smem + 32640;   // 8320
  float* H1 = smem + 16000;   // 16640 = 208x80, reuses Q+K region

  const int b = blockIdx.x;
  const int tid = threadIdx.x;
  const int nthr = blockDim.x;

  for (int i = tid; i < kN * kE; i += nthr) T[i] = tok[(long long)b * kN * kE + i];
  __syncthreads();

  const float rscale = 1.f / sqrtf((float)kE);

  for (int l = 0; l < kDepth; ++l) {
    // prefetch next layer's weights into cache while we compute this one
    if (l + 1 < kDepth && tid < 32) {
      __builtin_prefetch(wq + (l + 1) * kE * kE, 0, 1);
      __builtin_prefetch(wk + (l + 1) * kE * kE, 0, 1);
      __builtin_prefetch(wv + (l + 1) * kE * kE, 0, 1);
      __builtin_prefetch(fw1 + (l + 1) * kE * kFF, 0, 1);
      __builtin_prefetch(fw2 + (l + 1) * kFF * kE, 0, 1);
    }
    // ---- LN1 -> Y ----
    {
      const float* g = ln1_g + l * kE;
      const float* bb = ln1_b + l * kE;
      for (int row = tid; row < kN; row += nthr) {
        const float* tr = T + row * kE;
        float m = 0.f;
        for (int e = 0; e < kE; ++e) m += tr[e];
        m *= (1.f / kE);
        float v = 0.f;
        for (int e = 0; e < kE; ++e) { float d = tr[e] - m; v += d * d; }
        v *= (1.f / kE);
        float inv = rsqrtf(v + 1e-5f);
        float* yr = Y + row * kE;
        for (int e = 0; e < kE; ++e) yr[e] = (tr[e] - m) * inv * g[e] + bb[e];
      }
    }
    __syncthreads();
    // ---- Q,K,V projections (WMMA) ----
    wave_gemm(Y, kE, wq + l * kE * kE, kE, kE, bq + l * kE, Qm, kE, 13, kE, false, false, kNp);
    wave_gemm(Y, kE, wk + l * kE * kE, kE, kE, bk + l * kE, Km, kE, 13, kE, false, false, kNp);
    wave_gemm(Y, kE, wv + l * kE * kE, kE, kE, bvv + l * kE, Vm, kE, 13, kE, false, false, kNp);
    __syncthreads();
    // ---- windowed attention (|i-j|<=8), 5 heads of dim 8 -> O in Y ----
    for (int it = tid; it < kN * kNH; it += nthr) {
      int i = it / kNH, h = it % kNH;
      int j0 = i - kWS; if (j0 < 0) j0 = 0;
      int j1 = i + kWS; if (j1 > kN - 1) j1 = kN - 1;
      const float* qi = Qm + i * kE + h * 8;
      float w[2 * kWS + 1];
      float mx = -3.4e38f;
      for (int j = j0; j <= j1; ++j) {
        const float* kj = Km + j * kE + h * 8;
        float d = 0.f;
        for (int dd = 0; dd < 8; ++dd) d += qi[dd] * kj[dd];
        d *= rscale;
        w[j - j0] = d;
        mx = fmaxf(mx, d);
      }
      float z = 0.f;
      for (int j = j0; j <= j1; ++j) { float e = expf(w[j - j0] - mx); w[j - j0] = e; z += e; }
      float iz = 1.f / z;
      float o[8] = {0, 0, 0, 0, 0, 0, 0, 0};
      for (int j = j0; j <= j1; ++j) {
        const float* vj = Vm + j * kE + h * 8;
        float a = w[j - j0];
        for (int dd = 0; dd < 8; ++dd) o[dd] += a * vj[dd];
      }
      float* orow = Y + i * kE + h * 8;
      for (int dd = 0; dd < 8; ++dd) orow[dd] = o[dd] * iz;
    }
    __syncthreads();
    // ---- T += O @ Wo + bo (WMMA, accumulate into residual; T has only 200 rows -> row check) ----
    wave_gemm(Y, kE, wo + l * kE * kE, kE, kE, bo + l * kE, T, kE, 13, kE, true, true, kN);
    __syncthreads();
    // ---- LN2 -> Y ----
    {
      const float* g = ln2_g + l * kE;
      const float* bb = ln2_b + l * kE;
      for (int row = tid; row < kN; row += nthr) {
        const float* tr = T + row * kE;
        float m = 0.f;
        for (int e = 0; e < kE; ++e) m += tr[e];
        m *= (1.f / kE);
        float v = 0.f;
        for (int e = 0; e < kE; ++e) { float d = tr[e] - m; v += d * d; }
        v *= (1.f / kE);
        float inv = rsqrtf(v + 1e-5f);
        float* yr = Y + row * kE;
        for (int e = 0; e < kE; ++e) yr[e] = (tr[e] - m) * inv * g[e] + bb[e];
      }
    }
    __syncthreads();
    // ---- FFN (160 hidden) in two 80-wide chunks so the hidden fits in the Q/K region ----
    for (int c = 0; c < 2; ++c) {
      wave_gemm(Y, kE, fw1 + l * kE * kFF + c * 80, kFF, 80, fb1 + l * kFF + c * 80,
                H1, 80, 13, kE, false, false, kNp);
      __syncthreads();
      for (int i = tid; i < kN * 80; i += nthr) {
        float xv = H1[i];
        H1[i] = 0.5f * xv * (1.f + erff(xv * 0.70710678118654752f));  // exact GELU
      }
      __syncthreads();
      wave_gemm(H1, 80, fw2 + l * kFF * kE + c * 80 * kE, kE, kE,
                (c == 0) ? (fb2 + l * kE) : nullptr, T, kE, 13, 80, true, true, kN);
      __syncthreads();
    }
  }

  // ---- attention pooling head + classifier (tiny; scalar) ----
  float* s = Y;              // 200 scores
  float* pooled = Y + 208;   // 40
  float* hid = Y + 256;      // 64
  for (int i = tid; i < kN; i += nthr) {
    const float* tr = T + i * kE;
    float a = attn_b[0];
    for (int e = 0; e < kE; ++e) a += tr[e] * attn_w[e];
    s[i] = a;
  }
  __syncthreads();
  if (tid == 0) {
    float mx = -3.4e38f;
    for (int i = 0; i < kN; ++i) mx = fmaxf(mx, s[i]);
    float z = 0.f;
    for (int i = 0; i < kN; ++i) { float e = expf(s[i] - mx); s[i] = e; z += e; }
    Y[200] = 1.f / z;
  }
  __syncthreads();
  float iz = Y[200];
  for (int e = tid; e < kE; e += nthr) {
    float p = 0.f;
    for (int i = 0; i < kN; ++i) p += s[i] * T[i * kE + e];
    p *= iz;
    pooled[e] = p;
    out[b * kE + e] = p;  // pooled output
  }
  __syncthreads();
  for (int j = tid; j < kHid; j += nthr) {
    float h = cb1[j];
    for (int e = 0; e < kE; ++e) h += pooled[e] * cw1[e * kHid + j];
    hid[j] = elu1(h);
  }
  __syncthreads();
  for (int c = tid; c < kNC; c += nthr) {
    float lg = cb2[c];
    for (int j = 0; j < kHid; ++j) lg += hid[j] * cw2[j * kNC + c];
    out[kB * kE + b * kNC + c] = lg;  // logits after pooled block
  }
}

extern "C" void kernel_launch(void* const* d_in, const int* in_sizes, int n_in,
                              void* d_out, int out_size, void* d_ws, size_t ws_size,
                              hipStream_t stream) {
  (void)in_sizes; (void)n_in; (void)out_size; (void)ws_size;
  const float* x       = (const float*)d_in[0];
  const float* conv1_w = (const float*)d_in[1];
  const float* conv1_b = (const float*)d_in[2];
  const float* conv2_w = (const float*)d_in[3];
  const float* conv2_b = (const float*)d_in[4];
  const float* bn_g    = (const float*)d_in[5];
  const float* bn_b    = (const float*)d_in[6];
  const float* bn_m    = (const float*)d_in[7];
  const float* bn_v    = (const float*)d_in[8];
  const float* proj_w  = (const float*)d_in[9];
  const float* proj_b  = (const float*)d_in[10];
  const float* ln1_g   = (const float*)d_in[11];
  const float* ln1_b   = (const float*)d_in[12];
  const float* wq      = (const float*)d_in[13];
  const float* bq      = (const float*)d_in[14];
  const float* wk      = (const float*)d_in[15];
  const float* bk      = (const float*)d_in[16];
  const float* wv      = (const float*)d_in[17];
  const float* bv      = (const float*)d_in[18];
  const float* wo      = (const float*)d_in[19];
  const float* bo      = (const float*)d_in[20];
  const float* ln2_g   = (const float*)d_in[21];
  const float* ln2_b   = (const float*)d_in[22];
  const float* fw1     = (const float*)d_in[23];
  const float* fb1     = (const float*)d_in[24];
  const float* fw2     = (const float*)d_in[25];
  const float* fb2     = (const float*)d_in[26];
  const float* attn_w  = (const float*)d_in[27];
  const float* attn_b  = (const float*)d_in[28];
  const float* cw1     = (const float*)d_in[29];
  const float* cb1     = (const float*)d_in[30];
  const float* cw2     = (const float*)d_in[31];
  const float* cb2     = (const float*)d_in[32];

  float* out  = (float*)d_out;
  float* ws   = (float*)d_ws;
  float* weff = ws;                               // 22000 f32
  float* beff = ws + 22000;                       // 40 f32
  float* cact = ws + 22048;                       // 32*40*4055 f32 (~20.8 MB)
  float* tok  = cact + (size_t)kB * 40 * kCW;     // 32*200*40 f32 (~1 MB)

  k_fold<<<(22000 + 255) / 256, 256, 0, stream>>>(conv1_w, conv1_b, conv2_w, conv2_b,
                                                  weff, beff);
  long long tot = (long long)kB * 40 * kCW;
  k_conv<<<(int)((tot + 255) / 256), 256, 0, stream>>>(x, weff, beff, bn_g, bn_b, bn_m,
                                                       bn_v, cact);
  k_pool_proj<<<kB * kN, 64, 0, stream>>>(cact, proj_w, proj_b, tok);
  k_former<<<kB, 256, 0, stream>>>(tok, ln1_g, ln1_b, wq, bq, wk, bk, wv, bv, wo, bo,
                                   ln2_g, ln2_b, fw1, fb1, fw2, fb2, attn_w, attn_b,
                                   cw1, cb1, cw2, cb2, out);
}